// HybridQRC_89215060672685
// MI455X (gfx1250) — compile-verified
//
#include <hip/hip_runtime.h>
#include <math.h>

typedef __attribute__((ext_vector_type(2))) float v2f;
typedef __attribute__((ext_vector_type(8))) float v8f;

#define SW_     64
#define LW_     256
#define KSHORT_ 1024
#define KLONG_  2048
#define BLK_    64      // 2 waves of 32
#define SPB_    64      // samples per block
#define XS_STR  72      // padded row stride: 4 | 64 | 4
#define XL_STR  264     // padded row stride: 4 | 256 | 4

__global__ __launch_bounds__(BLK_) void hybrid_qrc_fused(
    const float* __restrict__ xs,  const float* __restrict__ xl,
    const float* __restrict__ csw, const float* __restrict__ csb,
    const float* __restrict__ lsw, const float* __restrict__ lsb,
    const float* __restrict__ clw, const float* __restrict__ clb,
    const float* __restrict__ llw, const float* __restrict__ llb,
    const float* __restrict__ qw,
    const float* __restrict__ f1w, const float* __restrict__ f1b,
    const float* __restrict__ f2w, const float* __restrict__ f2b,
    float* __restrict__ out)
{
  __shared__ __align__(16) float sh_xs[2][16 * XS_STR];  //  9 KB, zero-padded rows
  __shared__ __align__(16) float sh_xl[2][16 * XL_STR];  // 33 KB, zero-padded rows
  __shared__ float sh_f[SPB_][8];
  __shared__ float sh_gate[12][8];

  const int tid  = threadIdx.x;
  const int lane = tid & 31;
  const int wave = tid >> 5;
  const int block_base = blockIdx.x * SPB_;

  // ---- precompute the 12 Rot gate matrices (sample-independent) ----
  // Rot = [[e^{-i(phi+om)/2} c, -e^{+i(phi-om)/2} s],
  //        [e^{-i(phi-om)/2} s,  e^{+i(phi+om)/2} c]],  c=cos(th/2), s=sin(th/2)
  if (tid < 12) {
    const float phi = qw[tid * 3 + 0];
    const float th  = qw[tid * 3 + 1];
    const float om  = qw[tid * 3 + 2];
    float s2, c2, epi, epr, emi, emr;
    sincosf(0.5f * th, &s2, &c2);
    sincosf(-0.5f * (phi + om), &epi, &epr);
    sincosf(-0.5f * (phi - om), &emi, &emr);
    float* g = &sh_gate[tid][0];
    g[0] =  epr * c2;  g[1] =  epi * c2;   // M00
    g[2] = -emr * s2;  g[3] =  emi * s2;   // M01 = -conj(em)*s
    g[4] =  emr * s2;  g[5] =  emi * s2;   // M10
    g[6] =  epr * c2;  g[7] = -epi * c2;   // M11 = conj(ep)*c
  }

  // zero the guard pads of this wave's staged rows (done once; stores never touch pads)
  if (lane < 16) {
    float* r = &sh_xs[wave][lane * XS_STR];
    r[0] = r[1] = r[2] = r[3] = 0.0f;
    r[XS_STR - 4] = r[XS_STR - 3] = r[XS_STR - 2] = r[XS_STR - 1] = 0.0f;
    float* rl = &sh_xl[wave][lane * XL_STR];
    rl[0] = rl[1] = rl[2] = rl[3] = 0.0f;
    rl[XL_STR - 4] = rl[XL_STR - 3] = rl[XL_STR - 2] = rl[XL_STR - 1] = 0.0f;
  }

  const int   n     = lane & 15;          // WMMA column (feature) == A-row sample index
  const int   koff  = (lane >> 4) << 1;   // K offset per half-wave: 0 or 2
  const int   nrow  = (n < 6) ? n : 0;    // clamped weight row (loads always in-range)
  const float nmask = (n < 6) ? 1.0f : 0.0f;

  // ================= GEMM phase: two 16-sample tiles per wave =================
  for (int tile = 0; tile < 2; ++tile) {
    const int tile_local = wave * 32 + tile * 16;
    const int tile_base  = block_base + tile_local;

    { // stage tile inputs into padded LDS rows (float4, 16B-aligned row interiors)
      const float4* s4 = (const float4*)(xs + (size_t)tile_base * SW_);
#pragma unroll
      for (int k = 0; k < 8; ++k) {
        const int i = lane + k * 32;          // 256 float4 total
        const int s = i >> 4, c = i & 15;
        *(float4*)&sh_xs[wave][s * XS_STR + 4 + c * 4] = s4[i];
      }
      const float4* s4l = (const float4*)(xl + (size_t)tile_base * LW_);
#pragma unroll
      for (int k = 0; k < 32; ++k) {
        const int i = lane + k * 32;          // 1024 float4 total
        const int s = i >> 6, c = i & 63;
        *(float4*)&sh_xl[wave][s * XL_STR + 4 + c * 4] = s4l[i];
      }
    }
    __syncthreads();  // also publishes sh_gate / pad zeros on the first iteration

    const float* xrow_s = &sh_xs[wave][n * XS_STR + 4];
    const float* xrow_l = &sh_xl[wave][n * XL_STR + 4];

    // C init: bias columns (lsb+llb); the final *0.5 is applied at store
    v8f acc;
    {
      const float cinit = nmask * (lsb[nrow] + llb[nrow]);
#pragma unroll
      for (int r = 0; r < 8; ++r) acc[r] = cinit;
    }

    // ---- short branch: [16 x 1024] x [1024 x 6pad16], f32 16x16x4 WMMA ----
    // K index = ch*64 + position; channel weights are uniform -> SGPRs
    const float* bpS = lsw + nrow * KSHORT_ + koff;
#pragma unroll 1
    for (int ch = 0; ch < 16; ++ch) {
      const float w0 = csw[ch * 3 + 0];
      const float w1 = csw[ch * 3 + 1];
      const float w2 = csw[ch * 3 + 2];
      const float bs = csb[ch];
#pragma unroll 4
      for (int kq = 0; kq < 16; ++kq) {
        const int px = kq * 4 + koff;            // position of a.x (pads make taps safe)
        const float xm1 = xrow_s[px - 1];
        const float x0  = xrow_s[px];
        const float x1  = xrow_s[px + 1];
        const float x2  = xrow_s[px + 2];
        v2f a, b;
        a.x = fmaxf(fmaf(w2, x1, fmaf(w1, x0, fmaf(w0, xm1, bs))), 0.0f);
        a.y = fmaxf(fmaf(w2, x2, fmaf(w1, x1, fmaf(w0, x0,  bs))), 0.0f);
        const int kk = ch * 64 + kq * 4;
        b.x = nmask * bpS[kk];
        b.y = nmask * bpS[kk + 1];
        acc = __builtin_amdgcn_wmma_f32_16x16x4_f32(false, a, false, b,
                                                    (short)0, acc, false, false);
      }
    }

    // ---- long branch: [16 x 2048] x [2048 x 6pad16] ----
    // K index = ch*128 + pooled position q; conv pos p = 2q, maxpool over (p, p+1)
    const float* bpL = llw + nrow * KLONG_ + koff;
#pragma unroll 1
    for (int ch = 0; ch < 16; ++ch) {
      float wl[5];
#pragma unroll
      for (int t = 0; t < 5; ++t) wl[t] = clw[ch * 5 + t];
      const float bl = clb[ch];
#pragma unroll 4
      for (int kq = 0; kq < 32; ++kq) {
        const int q = kq * 4 + koff;             // pooled position of a.x
        const int p = 2 * q;
        float xv[8];                              // x[p-2 .. p+5], pads make it safe
#pragma unroll
        for (int t = 0; t < 8; ++t) xv[t] = xrow_l[p - 2 + t];
        float o0 = bl, o1 = bl, o2 = bl, o3 = bl;
#pragma unroll
        for (int t = 0; t < 5; ++t) {
          o0 = fmaf(wl[t], xv[t],     o0);       // conv(p)
          o1 = fmaf(wl[t], xv[t + 1], o1);       // conv(p+1)
          o2 = fmaf(wl[t], xv[t + 2], o2);       // conv(p+2)
          o3 = fmaf(wl[t], xv[t + 3], o3);       // conv(p+3)
        }
        v2f a, b;
        a.x = fmaxf(fmaxf(o0, o1), 0.0f);        // relu(max) == max(relu)
        a.y = fmaxf(fmaxf(o2, o3), 0.0f);
        const int kk = ch * 128 + kq * 4;
        b.x = nmask * bpL[kk];
        b.y = nmask * bpL[kk + 1];
        acc = __builtin_amdgcn_wmma_f32_16x16x4_f32(false, a, false, b,
                                                    (short)0, acc, false, false);
      }
    }

    // D layout: VGPR r -> (M=r, N=lane) lanes 0-15, (M=r+8, N=lane-16) lanes 16-31
    const int mbase = (lane >> 4) * 8;
    if (n < 6) {
#pragma unroll
      for (int r = 0; r < 8; ++r)
        sh_f[tile_local + mbase + r][n] = 0.5f * acc[r];
    }
    __syncthreads();
  }

  __syncthreads();

  // ================= quantum phase: one thread per sample =================
  float f[6];
#pragma unroll
  for (int j = 0; j < 6; ++j) f[j] = sh_f[tid][j];

  float re[64], im[64];
#pragma unroll
  for (int i = 0; i < 64; ++i) { re[i] = 0.0f; im[i] = 0.0f; }
  re[0] = 1.0f;

  // RY(pi*f_j) encoding: half-angle = pi*f/2; qubit j is bit (5-j)
  const float kHalfPi = 1.57079632679489661923f;
#pragma unroll
  for (int j = 0; j < 6; ++j) {
    float s, c;
    sincosf(kHalfPi * f[j], &s, &c);
    const int msk = 1 << (5 - j);
#pragma unroll
    for (int i = 0; i < 64; ++i) {
      if (i & msk) continue;
      const int i1 = i | msk;
      const float a0r = re[i],  a0i = im[i];
      const float a1r = re[i1], a1i = im[i1];
      re[i]  = c * a0r - s * a1r;  im[i]  = c * a0i - s * a1i;
      re[i1] = s * a0r + c * a1r;  im[i1] = s * a0i + c * a1i;
    }
  }

  // StronglyEntanglingLayers: Rot on each wire then CNOT ring with range r=l%5+1
#pragma unroll
  for (int l = 0; l < 2; ++l) {
#pragma unroll
    for (int j = 0; j < 6; ++j) {
      const float* g = &sh_gate[l * 6 + j][0];
      const float m00r = g[0], m00i = g[1], m01r = g[2], m01i = g[3];
      const float m10r = g[4], m10i = g[5], m11r = g[6], m11i = g[7];
      const int msk = 1 << (5 - j);
#pragma unroll
      for (int i = 0; i < 64; ++i) {
        if (i & msk) continue;
        const int i1 = i | msk;
        const float a0r = re[i],  a0i = im[i];
        const float a1r = re[i1], a1i = im[i1];
        re[i]  = m00r*a0r - m00i*a0i + m01r*a1r - m01i*a1i;
        im[i]  = m00r*a0i + m00i*a0r + m01r*a1i + m01i*a1r;
        re[i1] = m10r*a0r - m10i*a0i + m11r*a1r - m11i*a1i;
        im[i1] = m10r*a0i + m10i*a0r + m11r*a1i + m11i*a1r;
      }
    }
    const int rr = (l % 5) + 1;
#pragma unroll
    for (int j = 0; j < 6; ++j) {
      const int cm = 1 << (5 - j);
      const int tm = 1 << (5 - ((j + rr) % 6));
#pragma unroll
      for (int i = 0; i < 64; ++i) {
        if ((i & cm) && !(i & tm)) {   // control=1, target=0: swap with target=1
          const int i1 = i | tm;
          const float tr = re[i], ti = im[i];
          re[i] = re[i1];  im[i] = im[i1];
          re[i1] = tr;     im[i1] = ti;
        }
      }
    }
  }

  // <Z_k> expectations
  float z[6];
#pragma unroll
  for (int k = 0; k < 6; ++k) z[k] = 0.0f;
#pragma unroll
  for (int i = 0; i < 64; ++i) {
    const float p = re[i] * re[i] + im[i] * im[i];
#pragma unroll
    for (int k = 0; k < 6; ++k)
      z[k] += (i & (1 << (5 - k))) ? -p : p;
  }

  // fc1 -> relu -> fc2
  float o = f2b[0];
#pragma unroll
  for (int h = 0; h < 8; ++h) {
    float hv = f1b[h];
#pragma unroll
    for (int j = 0; j < 6; ++j) hv = fmaf(f1w[h * 6 + j], z[j], hv);
    hv = fmaxf(hv, 0.0f);
    o = fmaf(f2w[h], hv, o);
  }
  out[block_base + tid] = o;
}

extern "C" void kernel_launch(void* const* d_in, const int* in_sizes, int n_in,
                              void* d_out, int out_size, void* d_ws, size_t ws_size,
                              hipStream_t stream) {
  (void)n_in; (void)out_size; (void)d_ws; (void)ws_size;
  const float* xs  = (const float*)d_in[0];
  const float* xl  = (const float*)d_in[1];
  const float* csw = (const float*)d_in[2];
  const float* csb = (const float*)d_in[3];
  const float* lsw = (const float*)d_in[4];
  const float* lsb = (const float*)d_in[5];
  const float* clw = (const float*)d_in[6];
  const float* clb = (const float*)d_in[7];
  const float* llw = (const float*)d_in[8];
  const float* llb = (const float*)d_in[9];
  const float* qw  = (const float*)d_in[10];
  const float* f1w = (const float*)d_in[11];
  const float* f1b = (const float*)d_in[12];
  const float* f2w = (const float*)d_in[13];
  const float* f2b = (const float*)d_in[14];
  float* out = (float*)d_out;

  const int batch  = in_sizes[0] / SW_;   // 16384
  const int blocks = batch / SPB_;        // 256
  hybrid_qrc_fused<<<blocks, BLK_, 0, stream>>>(xs, xl, csw, csb, lsw, lsb,
                                                clw, clb, llw, llb, qw,
                                                f1w, f1b, f2w, f2b, out);
}